// DefendedModel_69733089017907
// MI455X (gfx1250) — compile-verified
//
#include <hip/hip_runtime.h>

typedef float v2f __attribute__((ext_vector_type(2)));
typedef float v8f __attribute__((ext_vector_type(8)));
typedef int   b64v __attribute__((vector_size(8)));   // matches builtin's param pointee
typedef __attribute__((address_space(1))) b64v* gptr_b64;
typedef __attribute__((address_space(3))) b64v* lptr_b64;

#define KDIM 3072
#define CDIM 10
#define KSEL 50
#define RPB  4       // rows per select-block
#define WPR  2       // waves per row (8 waves, 256 threads)
#define TILE 1024    // X points staged per LDS tile (40 KB per buffer)

#if __has_builtin(__builtin_amdgcn_global_load_async_to_lds_b64) && \
    __has_builtin(__builtin_amdgcn_s_wait_asynccnt)
#define USE_ASYNC_LDS 1
#else
#define USE_ASYNC_LDS 0
#endif

// ---------------------------------------------------------------------------
// Kernel 0: pad W (3072x10) into pair-interleaved zero-padded form:
// wpad[p*16 + n] = { W[2p][n], W[2p+1][n] } (n<10), else {0,0}.
// Makes the WMMA B-operand fetch a single unconditional 8B load.
// ---------------------------------------------------------------------------
__global__ __launch_bounds__(256)
void pad_w_kernel(const float* __restrict__ W, float2* __restrict__ wpad) {
  const int i = blockIdx.x * 256 + threadIdx.x;     // over (KDIM/2)*16
  if (i >= (KDIM / 2) * 16) return;
  const int p = i >> 4, n = i & 15;
  float2 v;
  v.x = (n < CDIM) ? W[(size_t)(2 * p) * CDIM + n] : 0.f;
  v.y = (n < CDIM) ? W[(size_t)(2 * p + 1) * CDIM + n] : 0.f;
  wpad[i] = v;
}

// ---------------------------------------------------------------------------
// Kernel 1: logits = x @ W + b via V_WMMA_F32_16X16X4_F32.
// One wave per block computes a 16x16 tile. Inner loop: 2x global_load_b64
// (A from x, B from padded W) + 1 WMMA, fully unconditional.
// ---------------------------------------------------------------------------
__global__ __launch_bounds__(32)
void logits_wmma_kernel(const float* __restrict__ x, const float2* __restrict__ wpad,
                        const float* __restrict__ bias, float* __restrict__ out,
                        float* __restrict__ wsLogits, unsigned* __restrict__ wsAbs) {
  __shared__ unsigned sAbs[16];
  const int lane = threadIdx.x;
  if (lane < 16) sAbs[lane] = 0u;
  __syncthreads();

  const int row0 = blockIdx.x * 16;
  const int m    = lane & 15;
  const int half = lane >> 4;        // 0 or 1
  const int n    = m;                // output column (C layout)
  const bool nv  = (n < CDIM);

  const v2f* arow = (const v2f*)(x + (size_t)(row0 + m) * KDIM + half * 2);
  const v2f* brow = (const v2f*)wpad;   // [p*16 + n]
  v8f c = {0.f, 0.f, 0.f, 0.f, 0.f, 0.f, 0.f, 0.f};

  for (int k = 0; k < KDIM; k += 4) {
    v2f a = arow[k >> 1];                         // x[(row0+m)*K + k + 2*half .. +1]
    v2f b = brow[((k >> 1) + half) * 16 + n];     // W rows k+2*half, k+2*half+1
    c = __builtin_amdgcn_wmma_f32_16x16x4_f32(false, a, false, b, (short)0, c,
                                              false, false);
  }

  const float bn = nv ? bias[n] : 0.f;
#pragma unroll
  for (int i = 0; i < 8; ++i) {
    const int mrow = i + half * 8;
    const int grow = row0 + mrow;
    float v = nv ? (c[i] + bn) : 0.f;
    wsLogits[grow * 16 + n] = v;
    if (nv) {
      out[grow * 11 + n] = v;
      atomicMax(&sAbs[mrow], __float_as_uint(fabsf(v)));   // ds_max_u32
    }
  }
  __syncthreads();
  if (lane < 16) wsAbs[row0 + lane] = sAbs[lane];
}

// ---------------------------------------------------------------------------
// Kernel 2: streaming exact top-K=50 per row, radix select on f32 bits of
// d2 = sum_c (logit_c - X_jc)^2. X is staged tile-by-tile into LDS (double
// buffered, async-to-LDS) and shared by all 8 waves, cutting L2 read traffic
// 8x versus per-wave global streaming.
// ---------------------------------------------------------------------------
__global__ __launch_bounds__(256)
void knn_select_kernel(const float* __restrict__ X, const int* __restrict__ Y,
                       const float* __restrict__ wsLogits,
                       const unsigned* __restrict__ wsAbs,
                       float* __restrict__ out, int N) {
  __shared__ float    sX[2][TILE * CDIM];    // 2 x 40 KB
  __shared__ float    sLg[RPB][CDIM];
  __shared__ unsigned sHist[RPB][256];
  __shared__ int      sTie[RPB];
  __shared__ int      sVotes[RPB];

  const int tid  = threadIdx.x;
  const int lane = tid & 31;
  const int wave = tid >> 5;
  const int rowL = wave / WPR;
  const int sub  = wave % WPR;
  const int ntiles = (N + TILE - 1) / TILE;

  if (tid < RPB * CDIM)
    sLg[tid / CDIM][tid % CDIM] =
        wsLogits[(size_t)(blockIdx.x * RPB + tid / CDIM) * 16 + (tid % CDIM)];
  if (tid < RPB) { sTie[tid] = 0; sVotes[tid] = 0; }
  __syncthreads();

  float lg[CDIM];
#pragma unroll
  for (int c2 = 0; c2 < CDIM; ++c2) lg[c2] = sLg[rowL][c2];

  const int stride = 32 * WPR;   // 64 lanes cooperate per row
  const int start  = sub * 32 + lane;

  // --- tile staging (cooperative across all 256 threads) -------------------
  auto stage_issue = [&](int t, int buf) {
    const int j0  = t * TILE;
    const int nf2 = ((N - j0) < TILE ? (N - j0) : TILE) * (CDIM / 2);
    const float2* src = (const float2*)(X + (size_t)j0 * CDIM);
    float2* dst = (float2*)sX[buf];
    for (int i = tid; i < nf2; i += 256) {
#if USE_ASYNC_LDS
      __builtin_amdgcn_global_load_async_to_lds_b64(
          (gptr_b64)(src + i), (lptr_b64)(dst + i), 0, 0);
#else
      dst[i] = src[i];
#endif
    }
  };
  auto stage_wait = [&]() {
#if USE_ASYNC_LDS
    __builtin_amdgcn_s_wait_asynccnt(0);
#endif
  };

  auto dist_key = [&](const float* base, int jl) -> unsigned {
    const float2* xp = (const float2*)(base + (size_t)jl * CDIM);
    float d2 = 0.f;
#pragma unroll
    for (int c2 = 0; c2 < CDIM / 2; ++c2) {
      float2 xv = xp[c2];
      float df0 = xv.x - lg[2 * c2];
      float df1 = xv.y - lg[2 * c2 + 1];
      d2 = fmaf(df0, df0, d2);
      d2 = fmaf(df1, df1, d2);
    }
    return __float_as_uint(d2);
  };

  unsigned prefix = 0u, himask = 0u;
  int krem = KSEL;

  // --- 4-pass radix select --------------------------------------------------
  for (int byt = 3; byt >= 0; --byt) {
    for (int b2 = start; b2 < 256; b2 += stride) sHist[rowL][b2] = 0u;
    stage_issue(0, 0);
    stage_wait();
    __syncthreads();
    const unsigned shift = (unsigned)(byt * 8);

    for (int t = 0; t < ntiles; ++t) {
      const int buf  = t & 1;
      const int npts = (N - t * TILE) < TILE ? (N - t * TILE) : TILE;
      if (t + 1 < ntiles) stage_issue(t + 1, buf ^ 1);
      for (int jl = start; jl < npts; jl += stride) {
        const unsigned key = dist_key(sX[buf], jl);
        if ((key & himask) == prefix)
          atomicAdd(&sHist[rowL][(key >> shift) & 255u], 1u);
      }
      stage_wait();
      __syncthreads();
    }

    // per-wave redundant bin selection (identical across a row's waves)
    unsigned cnt[8]; unsigned s = 0u;
#pragma unroll
    for (int i = 0; i < 8; ++i) { cnt[i] = sHist[rowL][lane * 8 + i]; s += cnt[i]; }
    unsigned inc = s;
#pragma unroll
    for (int d = 1; d < 32; d <<= 1) {
      unsigned tt = __shfl_up(inc, d, 32);
      if (lane >= d) inc += tt;
    }
    const unsigned exc = inc - s;
    const bool hit = (exc < (unsigned)krem) && (inc >= (unsigned)krem);
    unsigned long long bm = __ballot(hit);
    const int hitlane = bm ? (__ffsll(bm) - 1) : 0;

    unsigned binSel = 0u, below = 0u;
    if (lane == hitlane) {
      unsigned run = exc;
      for (int i = 0; i < 8; ++i) {
        if (run + cnt[i] >= (unsigned)krem) { binSel = (unsigned)(lane * 8 + i); below = run; break; }
        run += cnt[i];
      }
    }
    binSel = __shfl(binSel, hitlane, 32);
    below  = __shfl(below, hitlane, 32);

    krem   -= (int)below;
    prefix |= (binSel << shift);
    himask |= (0xFFu << shift);
    __syncthreads();
  }

  // --- final vote scan under exact k-th key T -------------------------------
  const unsigned T = prefix;
  int votes = 0;
  stage_issue(0, 0);
  stage_wait();
  __syncthreads();
  for (int t = 0; t < ntiles; ++t) {
    const int buf  = t & 1;
    const int j0   = t * TILE;
    const int npts = (N - j0) < TILE ? (N - j0) : TILE;
    if (t + 1 < ntiles) stage_issue(t + 1, buf ^ 1);
    for (int jl = start; jl < npts; jl += stride) {
      const unsigned key = dist_key(sX[buf], jl);
      if (key < T) {
        votes += 2 * Y[j0 + jl] - 1;
      } else if (key == T) {
        int old = atomicAdd(&sTie[rowL], 1);
        if (old < krem) votes += 2 * Y[j0 + jl] - 1;
      }
    }
    stage_wait();
    __syncthreads();
  }
#pragma unroll
  for (int d = 16; d > 0; d >>= 1) votes += __shfl_down(votes, d, 32);
  if (lane == 0) atomicAdd(&sVotes[rowL], votes);
  __syncthreads();

  if (tid < RPB) {
    const int v = sVotes[tid];
    const float sv = (v > 0) ? 1.f : ((v < 0) ? -1.f : 0.f);
    const float am = __uint_as_float(wsAbs[blockIdx.x * RPB + tid]);
    out[(size_t)(blockIdx.x * RPB + tid) * 11 + 10] = 2.f * sv * am;
  }
}

// ---------------------------------------------------------------------------
extern "C" void kernel_launch(void* const* d_in, const int* in_sizes, int n_in,
                              void* d_out, int out_size, void* d_ws, size_t ws_size,
                              hipStream_t stream) {
  const float* x    = (const float*)d_in[0];   // (512, 3072)
  const float* W    = (const float*)d_in[1];   // (3072, 10)
  const float* bias = (const float*)d_in[2];   // (10,)
  const float* X    = (const float*)d_in[3];   // (100000, 10)
  const int*   Y    = (const int*)d_in[4];     // (100000,)

  const int Bq = in_sizes[0] / KDIM;           // 512 rows
  const int N  = in_sizes[3] / CDIM;           // 100000 points

  float* out      = (float*)d_out;             // (512, 11)
  float* wsLogits = (float*)d_ws;              // Bq*16 floats
  unsigned* wsAbs = (unsigned*)(wsLogits + (size_t)Bq * 16);   // Bq uints
  float2* wsWpad  = (float2*)(wsLogits + (size_t)Bq * 17);     // (KDIM/2)*16 float2, 8B aligned

  const int wcount = (KDIM / 2) * 16;
  pad_w_kernel<<<(wcount + 255) / 256, 256, 0, stream>>>(W, wsWpad);
  logits_wmma_kernel<<<Bq / 16, 32, 0, stream>>>(x, wsWpad, bias, out, wsLogits, wsAbs);
  knn_select_kernel<<<Bq / RPB, 256, 0, stream>>>(X, Y, wsLogits, wsAbs, out, N);
}